// MaskASPPDeformable_52381421142267
// MI455X (gfx1250) — compile-verified
//
#include <hip/hip_runtime.h>
#include <math.h>

typedef float v2f __attribute__((ext_vector_type(2)));
typedef float v8f __attribute__((ext_vector_type(8)));

#define B_    2
#define C_    32
#define H_    128
#define W_    128
#define HW_   (H_ * W_)
#define NPIX_ (B_ * HW_)

static __device__ __forceinline__ v8f wmma_f32(v2f a, v2f b, v8f c) {
  // V_WMMA_F32_16X16X4_F32: D = A(16x4) x B(4x16) + C(16x16), fp32 throughout.
  return __builtin_amdgcn_wmma_f32_16x16x4_f32(false, a, false, b, (short)0, c,
                                               false, false);
}

static __device__ __forceinline__ float gelu_exact(float x) {
  return 0.5f * x * (1.0f + erff(x * 0.70710678118654752f));
}

// ---------------------------------------------------------------------------
// Weight repack: OIHW -> [tap*Cin + ci][Npad] (zero-padded over output chans)
// ---------------------------------------------------------------------------
__global__ void k_repack_conv_w(const float* __restrict__ src, float* __restrict__ dst,
                                int O, int Cin, int ks, int Npad) {
  int K = ks * ks;
  int tot = K * Cin * Npad;
  int i = blockIdx.x * blockDim.x + threadIdx.x;
  if (i >= tot) return;
  int o = i % Npad;
  int kc = i / Npad;          // tap*Cin + ci
  int ci = kc % Cin;
  int tap = kc / Cin;
  int ky = tap / ks, kx = tap % ks;
  float v = 0.0f;
  if (o < O) v = src[((size_t)(o * Cin + ci) * ks + ky) * ks + kx];
  dst[i] = v;
}

// Deform weight repack: w[o][c][k] (OIHW flattened taps) -> [k*Cin + c][32]
__global__ void k_repack_deform_w(const float* __restrict__ src, float* __restrict__ dst,
                                  int Cin, int ks) {
  int K = ks * ks;
  int tot = K * Cin * 32;
  int i = blockIdx.x * blockDim.x + threadIdx.x;
  if (i >= tot) return;
  int o = i % 32;
  int kc = i / 32;            // k*Cin + c
  int c = kc % Cin;
  int k = kc / Cin;
  int ky = k / ks, kx = k % ks;
  dst[i] = src[((size_t)(o * Cin + c) * ks + ky) * ks + kx];
}

// ---------------------------------------------------------------------------
// Stage 1: concat(x, masks) -> 1x1 conv(33->32) + bias -> LN -> GELU, NHWC out
// ---------------------------------------------------------------------------
__global__ void k_stage1(const float* __restrict__ x, const float* __restrict__ masks,
                         const float* __restrict__ w, const float* __restrict__ bias,
                         const float* __restrict__ g, const float* __restrict__ be,
                         float* __restrict__ h0) {
  int p = blockIdx.x * blockDim.x + threadIdx.x;
  if (p >= NPIX_) return;
  int b = p / HW_;
  int s = p % HW_;
  float in[33];
#pragma unroll
  for (int c = 0; c < 32; ++c) in[c] = x[(size_t)(b * 32 + c) * HW_ + s];
  in[32] = masks[(size_t)b * HW_ + s];
  float o[32];
  float mu = 0.0f;
  for (int oc = 0; oc < 32; ++oc) {
    float acc = bias[oc];
    const float* wr = w + oc * 33;
#pragma unroll
    for (int c = 0; c < 33; ++c) acc += wr[c] * in[c];
    o[oc] = acc;
    mu += acc;
  }
  mu *= (1.0f / 32.0f);
  float var = 0.0f;
#pragma unroll
  for (int oc = 0; oc < 32; ++oc) { float d = o[oc] - mu; var += d * d; }
  var *= (1.0f / 32.0f);
  float rs = rsqrtf(var + 1e-6f);
#pragma unroll
  for (int oc = 0; oc < 32; ++oc)
    h0[(size_t)p * 32 + oc] = gelu_exact((o[oc] - mu) * rs * g[oc] + be[oc]);
}

// ---------------------------------------------------------------------------
// Per-pixel LayerNorm(32ch) + GELU. nchw!=0 -> write NCHW (final output).
// ---------------------------------------------------------------------------
__global__ void k_ln_gelu(const float* __restrict__ in, const float* __restrict__ g,
                          const float* __restrict__ be, float* __restrict__ out,
                          int out_stride, int nchw) {
  int p = blockIdx.x * blockDim.x + threadIdx.x;
  if (p >= NPIX_) return;
  float v[32];
  float mu = 0.0f;
#pragma unroll
  for (int c = 0; c < 32; ++c) { v[c] = in[(size_t)p * 32 + c]; mu += v[c]; }
  mu *= (1.0f / 32.0f);
  float var = 0.0f;
#pragma unroll
  for (int c = 0; c < 32; ++c) { float d = v[c] - mu; var += d * d; }
  var *= (1.0f / 32.0f);
  float rs = rsqrtf(var + 1e-6f);
  if (nchw) {
    int b = p / HW_, s = p % HW_;
#pragma unroll
    for (int c = 0; c < 32; ++c)
      out[(size_t)(b * 32 + c) * HW_ + s] = gelu_exact((v[c] - mu) * rs * g[c] + be[c]);
  } else {
#pragma unroll
    for (int c = 0; c < 32; ++c)
      out[(size_t)p * out_stride + c] = gelu_exact((v[c] - mu) * rs * g[c] + be[c]);
  }
}

// ---------------------------------------------------------------------------
// Generic conv-as-GEMM via implicit im2col + V_WMMA_F32_16X16X4_F32.
// Input NHWC with pixel stride in_stride (>= Cin). Weight packed [K*Cin][Npad].
// epi: 0 none, 1 +bias, 2 2*sigmoid(x+bias). Output [pix][Oreal].
// 4 waves/block: each wave a 16-pixel tile, all share n-tile -> per-tap B
// block (Cin x 16) staged once per block into LDS (padded pitch 17).
// ---------------------------------------------------------------------------
__global__ void k_conv_wmma(const float* __restrict__ in, int in_stride, int Cin,
                            int ks, int pad, const float* __restrict__ Wp, int Npad,
                            const float* __restrict__ bias, int Oreal,
                            float* __restrict__ out, int epi) {
  __shared__ float Bs[96 * 17];  // max Cin=96, pitch 17 to dodge bank conflicts
  int lane = threadIdx.x & 31;
  int wv = threadIdx.x >> 5;
  int ptile = blockIdx.x * 4 + wv;
  int n0 = blockIdx.y * 16;
  int p0 = ptile * 16;
  int b = p0 / HW_;
  int rem = p0 % HW_;
  int y = rem / W_;
  int x0 = rem % W_;
  int m = lane & 15;   // doubles as: pixel-in-tile for A, column n for B/D
  int kh = lane >> 4;  // K-half selector (fp32 WMMA A/B layout)
  int xm = x0 + m;

  v8f acc = {0.f, 0.f, 0.f, 0.f, 0.f, 0.f, 0.f, 0.f};
  const int cgroups = Cin >> 2;
  const int nstage = Cin * 16;
  for (int ky = 0; ky < ks; ++ky) {
    int yy = y + ky - pad;
    bool yok = (yy >= 0) && (yy < H_);
    for (int kx = 0; kx < ks; ++kx) {
      int tap = ky * ks + kx;
      // cooperative stage of this tap's B block into LDS
      const float* Bg = Wp + (size_t)(tap * Cin) * Npad + n0;
      for (int t = threadIdx.x; t < nstage; t += 128) {
        int r = t >> 4, n = t & 15;
        Bs[r * 17 + n] = Bg[(size_t)r * Npad + n];
      }
      __syncthreads();

      int xx = xm + kx - pad;
      bool ok = yok && (xx >= 0) && (xx < W_);
      long pixoff = ((long)(b * H_ + yy) * W_ + xx) * in_stride;
      for (int cg = 0; cg < cgroups; ++cg) {
        int ci = cg * 4 + 2 * kh;
        v2f a;
        if (ok) a = *(const v2f*)(in + pixoff + ci);
        else    { a.x = 0.0f; a.y = 0.0f; }
        v2f bb;
        bb.x = Bs[ci * 17 + m];
        bb.y = Bs[(ci + 1) * 17 + m];
        acc = wmma_f32(a, bb, acc);
      }
      __syncthreads();
    }
  }
  int col = n0 + m;
  if (col < Oreal) {
    float bval = (epi >= 1) ? bias[col] : 0.0f;
#pragma unroll
    for (int r = 0; r < 8; ++r) {
      int mr = r + 8 * kh;
      float v = acc[r] + bval;
      if (epi == 2) v = 2.0f / (1.0f + expf(-v));
      out[(size_t)(p0 + mr) * Oreal + col] = v;
    }
  }
}

// ---------------------------------------------------------------------------
// Fused modulated-deformable conv: bilinear sample (zero-pad semantics) with
// modulator folded into corner weights, fed straight into WMMA fp32 GEMM
// against packed regular_conv weight [k*32 + c][32] staged per-tap in LDS.
// Output [pix][32].
// ---------------------------------------------------------------------------
__global__ void k_deform_wmma(const float* __restrict__ h, int h_stride,
                              const float* __restrict__ off,
                              const float* __restrict__ modv,
                              const float* __restrict__ Wd, int ks, int pad,
                              float* __restrict__ out) {
  __shared__ float Ws[32 * 33];  // one tap: 32 chans x 32 outs, pitch 33
  int K = ks * ks;
  int lane = threadIdx.x & 31;
  int wv = threadIdx.x >> 5;
  int ptile = blockIdx.x * 4 + wv;
  int p0 = ptile * 16;
  int b = p0 / HW_;
  int rem = p0 % HW_;
  int y = rem / W_;
  int x0 = rem % W_;
  int m = lane & 15;
  int kh = lane >> 4;
  int p = p0 + m;
  int x = x0 + m;

  v8f c0 = {0.f, 0.f, 0.f, 0.f, 0.f, 0.f, 0.f, 0.f};
  v8f c1 = {0.f, 0.f, 0.f, 0.f, 0.f, 0.f, 0.f, 0.f};

  for (int k = 0; k < K; ++k) {
    // cooperative stage of this tap's weight block
    const float* Wg = Wd + (size_t)(k * 32) * 32;
    for (int t = threadIdx.x; t < 1024; t += 128) {
      int r = t >> 5, oc = t & 31;
      Ws[r * 33 + oc] = Wg[(size_t)r * 32 + oc];
    }
    __syncthreads();

    int ky = k / ks, kx = k % ks;
    float oy = off[(size_t)p * (2 * K) + 2 * k];
    float ox = off[(size_t)p * (2 * K) + 2 * k + 1];
    float mk = modv[(size_t)p * K + k];
    float py = (float)(y + ky - pad) + oy;
    float px = (float)(x + kx - pad) + ox;
    float fy = floorf(py), fx = floorf(px);
    float wy = py - fy, wx = px - fx;
    int iy0 = (int)fy, ix0 = (int)fx;
    int iy1 = iy0 + 1, ix1 = ix0 + 1;
    bool y0ok = (iy0 >= 0) && (iy0 < H_);
    bool y1ok = (iy1 >= 0) && (iy1 < H_);
    bool x0ok = (ix0 >= 0) && (ix0 < W_);
    bool x1ok = (ix1 >= 0) && (ix1 < W_);
    float w00 = (y0ok && x0ok) ? (1.0f - wy) * (1.0f - wx) * mk : 0.0f;
    float w01 = (y0ok && x1ok) ? (1.0f - wy) * wx * mk : 0.0f;
    float w10 = (y1ok && x0ok) ? wy * (1.0f - wx) * mk : 0.0f;
    float w11 = (y1ok && x1ok) ? wy * wx * mk : 0.0f;
    int cy0 = min(max(iy0, 0), H_ - 1), cy1 = min(max(iy1, 0), H_ - 1);
    int cx0 = min(max(ix0, 0), W_ - 1), cx1 = min(max(ix1, 0), W_ - 1);
    const float* p00 = h + ((size_t)(b * H_ + cy0) * W_ + cx0) * h_stride;
    const float* p01 = h + ((size_t)(b * H_ + cy0) * W_ + cx1) * h_stride;
    const float* p10 = h + ((size_t)(b * H_ + cy1) * W_ + cx0) * h_stride;
    const float* p11 = h + ((size_t)(b * H_ + cy1) * W_ + cx1) * h_stride;
#pragma unroll
    for (int s = 0; s < 8; ++s) {
      int ci = 4 * s + 2 * kh;
      v2f q00 = *(const v2f*)(p00 + ci);
      v2f q01 = *(const v2f*)(p01 + ci);
      v2f q10 = *(const v2f*)(p10 + ci);
      v2f q11 = *(const v2f*)(p11 + ci);
      v2f a;
      a.x = w00 * q00.x + w01 * q01.x + w10 * q10.x + w11 * q11.x;
      a.y = w00 * q00.y + w01 * q01.y + w10 * q10.y + w11 * q11.y;
      v2f b0, b1;
      b0.x = Ws[ci * 33 + m];
      b0.y = Ws[(ci + 1) * 33 + m];
      b1.x = Ws[ci * 33 + 16 + m];
      b1.y = Ws[(ci + 1) * 33 + 16 + m];
      c0 = wmma_f32(a, b0, c0);
      c1 = wmma_f32(a, b1, c1);
    }
    __syncthreads();
  }
#pragma unroll
  for (int r = 0; r < 8; ++r) {
    int mr = r + 8 * kh;
    out[(size_t)(p0 + mr) * 32 + m] = c0[r];
    out[(size_t)(p0 + mr) * 32 + 16 + m] = c1[r];
  }
}

// ---------------------------------------------------------------------------
extern "C" void kernel_launch(void* const* d_in, const int* in_sizes, int n_in,
                              void* d_out, int out_size, void* d_ws, size_t ws_size,
                              hipStream_t stream) {
  (void)in_sizes; (void)n_in; (void)out_size; (void)ws_size;
  const float* x      = (const float*)d_in[0];
  const float* masks  = (const float*)d_in[1];
  const float* me_w1  = (const float*)d_in[2];
  const float* me_b1  = (const float*)d_in[3];
  const float* me_g1  = (const float*)d_in[4];
  const float* me_be1 = (const float*)d_in[5];
  const float* me_w2  = (const float*)d_in[6];
  const float* me_b2  = (const float*)d_in[7];
  const float* me_g2  = (const float*)d_in[8];
  const float* me_be2 = (const float*)d_in[9];
  const float* a1_ow  = (const float*)d_in[10];
  const float* a1_ob  = (const float*)d_in[11];
  const float* a1_mw  = (const float*)d_in[12];
  const float* a1_mb  = (const float*)d_in[13];
  const float* a1_rw  = (const float*)d_in[14];
  const float* a1_g   = (const float*)d_in[15];
  const float* a1_b   = (const float*)d_in[16];
  const float* a2_ow  = (const float*)d_in[17];
  const float* a2_ob  = (const float*)d_in[18];
  const float* a2_mw  = (const float*)d_in[19];
  const float* a2_mb  = (const float*)d_in[20];
  const float* a2_rw  = (const float*)d_in[21];
  const float* a2_g   = (const float*)d_in[22];
  const float* a2_b   = (const float*)d_in[23];
  const float* co_w   = (const float*)d_in[24];
  const float* co_g   = (const float*)d_in[25];
  const float* co_b   = (const float*)d_in[26];

  float* ws = (float*)d_ws;
  size_t o = 0;
  float* h0   = ws + o; o += (size_t)NPIX_ * 32;
  float* ytmp = ws + o; o += (size_t)NPIX_ * 32;
  float* hcat = ws + o; o += (size_t)NPIX_ * 96;
  float* offb = ws + o; o += (size_t)NPIX_ * 98;
  float* modb = ws + o; o += (size_t)NPIX_ * 49;
  float* w_me2  = ws + o; o += 288 * 32;
  float* w_a1ow = ws + o; o += 288 * 32;
  float* w_a1mw = ws + o; o += 288 * 16;
  float* w_a1rw = ws + o; o += 288 * 32;
  float* w_a2ow = ws + o; o += (size_t)1568 * 112;
  float* w_a2mw = ws + o; o += (size_t)1568 * 64;
  float* w_a2rw = ws + o; o += (size_t)1568 * 32;
  float* w_co   = ws + o; o += 96 * 32;

  auto blks = [](int tot) { return (tot + 255) / 256; };

  // ---- weight repacks ----
  k_repack_conv_w<<<blks(9 * 32 * 32), 256, 0, stream>>>(me_w2, w_me2, 32, 32, 3, 32);
  k_repack_conv_w<<<blks(9 * 32 * 32), 256, 0, stream>>>(a1_ow, w_a1ow, 18, 32, 3, 32);
  k_repack_conv_w<<<blks(9 * 32 * 16), 256, 0, stream>>>(a1_mw, w_a1mw, 9, 32, 3, 16);
  k_repack_conv_w<<<blks(49 * 32 * 112), 256, 0, stream>>>(a2_ow, w_a2ow, 98, 32, 7, 112);
  k_repack_conv_w<<<blks(49 * 32 * 64), 256, 0, stream>>>(a2_mw, w_a2mw, 49, 32, 7, 64);
  k_repack_conv_w<<<blks(1 * 96 * 32), 256, 0, stream>>>(co_w, w_co, 32, 96, 1, 32);
  k_repack_deform_w<<<blks(9 * 32 * 32), 256, 0, stream>>>(a1_rw, w_a1rw, 32, 3);
  k_repack_deform_w<<<blks(49 * 32 * 32), 256, 0, stream>>>(a2_rw, w_a2rw, 32, 7);

  const int LNB = blks(NPIX_);
  const int GX = NPIX_ / 64;  // 16-pixel tiles, 4 waves/block (exact cover)

  // ---- mask embedding ----
  k_stage1<<<LNB, 256, 0, stream>>>(x, masks, me_w1, me_b1, me_g1, me_be1, h0);
  k_conv_wmma<<<dim3(GX, 2), 128, 0, stream>>>(h0, 32, 32, 3, 1, w_me2, 32, me_b2, 32, ytmp, 1);
  k_ln_gelu<<<LNB, 256, 0, stream>>>(ytmp, me_g2, me_be2, hcat + 0, 96, 0);

  // ---- branch 1: 3x3 deformable ----
  k_conv_wmma<<<dim3(GX, 2), 128, 0, stream>>>(hcat, 96, 32, 3, 1, w_a1ow, 32, a1_ob, 18, offb, 1);
  k_conv_wmma<<<dim3(GX, 1), 128, 0, stream>>>(hcat, 96, 32, 3, 1, w_a1mw, 16, a1_mb, 9, modb, 2);
  k_deform_wmma<<<GX, 128, 0, stream>>>(hcat, 96, offb, modb, w_a1rw, 3, 1, ytmp);
  k_ln_gelu<<<LNB, 256, 0, stream>>>(ytmp, a1_g, a1_b, hcat + 32, 96, 0);

  // ---- branch 2: 7x7 deformable ----
  k_conv_wmma<<<dim3(GX, 7), 128, 0, stream>>>(hcat, 96, 32, 7, 3, w_a2ow, 112, a2_ob, 98, offb, 1);
  k_conv_wmma<<<dim3(GX, 4), 128, 0, stream>>>(hcat, 96, 32, 7, 3, w_a2mw, 64, a2_mb, 49, modb, 2);
  k_deform_wmma<<<GX, 128, 0, stream>>>(hcat, 96, offb, modb, w_a2rw, 7, 3, ytmp);
  k_ln_gelu<<<LNB, 256, 0, stream>>>(ytmp, a2_g, a2_b, hcat + 64, 96, 0);

  // ---- output projection ----
  k_conv_wmma<<<dim3(GX, 2), 128, 0, stream>>>(hcat, 96, 96, 1, 0, w_co, 32, nullptr, 32, ytmp, 0);
  k_ln_gelu<<<LNB, 256, 0, stream>>>(ytmp, co_g, co_b, (float*)d_out, 0, 1);
}